// SegmentalLoss_5892695130367
// MI455X (gfx1250) — compile-verified
//
#include <hip/hip_runtime.h>
#include <stdint.h>

// ---------------- problem constants ----------------
#define B_      32
#define T_      480000
#define SEG_    1024
#define HOP_    256
#define PAD_    512
#define TP_     (T_ + 2 * PAD_)          // 481024 padded length
#define F_      1876                     // frames per batch
#define EPS_    1e-8f

// ---------------- tiling ----------------
#define FPW     13                       // frames per wave tile (blocks used: 16)
#define WPW     8                        // waves per workgroup (256 threads, wave32)
#define FPG     (FPW * WPW)              // 104 frames per WG
#define WGPB    ((F_ + FPG - 1) / FPG)   // 19 WGs per batch
#define NBLK_W  16                       // 256-sample blocks per wave tile
#define ROWP    258                      // LDS row stride in floats (bank padding, even for b64)
#define WAVE_F  (2 * NBLK_W * ROWP)      // floats per wave region (e + t)
#define AFRAG_N (64 * 32)                // A-fragment table entries (float2 each)

typedef __attribute__((ext_vector_type(2))) float v2f;
typedef __attribute__((ext_vector_type(8))) float v8f;

// ---- CDNA5 async global->LDS copy (per-lane gather, per-lane LDS dest) ----
__device__ __forceinline__ void async_copy_b32(uint32_t lds_byte_addr,
                                               unsigned long long gaddr) {
  asm volatile("global_load_async_to_lds_b32 %0, %1, off"
               :
               : "v"(lds_byte_addr), "v"(gaddr)
               : "memory");
}

__device__ __forceinline__ void wait_async0() {
#if __has_builtin(__builtin_amdgcn_s_wait_asynccnt)
  __builtin_amdgcn_s_wait_asynccnt(0);
#else
  asm volatile("s_wait_asynccnt 0" ::: "memory");
#endif
}

// reflect-pad index map: padded index i -> source index in [0, T)
__device__ __forceinline__ int reflect_idx(long i) {
  long j = i - PAD_;
  if (j < 0) j = -j;
  if (j >= T_) j = 2L * (T_ - 1) - j;
  if (j < 0) j = 0;                 // safety clamp for tail-tile overread
  if (j >= T_) j = T_ - 1;
  return (int)j;
}

// squared-squared Hann: w2(k) = hann(k)^2, hann(k) = sin^2(pi k / SEG)
__device__ __forceinline__ float w2val(int k) {
  float s = sinf(3.14159265358979323846f * (float)k / (float)SEG_);
  float h = s * s;
  return h * h;
}

// ---------------- kernel 1: per-batch means ----------------
extern "C" __global__ void __launch_bounds__(256)
segsisdr_means(const float* __restrict__ est, const float* __restrict__ tgt,
               float* __restrict__ means) {
  __shared__ float red[2][256];
  const int b = blockIdx.x, tid = threadIdx.x;
  const float* eg = est + (size_t)b * T_;
  const float* tg = tgt + (size_t)b * T_;
  float se = 0.f, st = 0.f;
  for (int i = tid; i < T_; i += 256) { se += eg[i]; st += tg[i]; }
  red[0][tid] = se; red[1][tid] = st;
  __syncthreads();
  for (int s = 128; s > 0; s >>= 1) {
    if (tid < s) { red[0][tid] += red[0][tid + s]; red[1][tid] += red[1][tid + s]; }
    __syncthreads();
  }
  if (tid == 0) {
    means[2 * b + 0] = red[0][0] / (float)T_;
    means[2 * b + 1] = red[1][0] / (float)T_;
  }
}

// ---------------- kernel 2: segmental SI-SDR via WMMA taps-GEMM ----------------
// Per wave: 16 blocks of 256 padded samples staged to LDS (async), then
//   P_s[j, n] = sum_r w2[256j + r] * X_s[block n, r]   via V_WMMA_F32_16X16X4_F32
// with A = w2 reshaped 4x256 (rows 0..3 of M, rows 4..15 zero), B columns = 16
// blocks of X_s, for the three product signals s in {e*t, t*t, e*e}.
// Frame result: out[f] = sum_j P[j, f + j]   (diagonal gather with __shfl).
extern "C" __global__ void __launch_bounds__(256)
segsisdr_main(const float* __restrict__ est, const float* __restrict__ tgt,
              const float* __restrict__ means, float* __restrict__ out) {
  extern __shared__ float lds[];
  v2f*   Afrag = (v2f*)lds;                    // 2048 float2 = 16 KB: per-(kk,lane) A fragments
  float* wbase = lds + 2 * AFRAG_N;            // per-wave staging regions

  const int tid  = threadIdx.x;
  const int lane = tid & 31;
  const int wave = tid >> 5;
  const int b    = blockIdx.x / WGPB;
  const int wgi  = blockIdx.x % WGPB;

  // Build zero-padded A-fragment table once per WG, laid out exactly as the
  // V_WMMA_F32_16X16X4_F32 A operand wants it per lane (ISA 7.12.2):
  //   v0 = A[m][4kk + kh], v1 = A[m][4kk + kh + 1], m = lane&15, kh = (lane>>4)*2
  // with A[m][k] = w2(m*256 + k) for m < 4 (taps), else 0.
  for (int idx = tid; idx < AFRAG_N; idx += 256) {
    const int kk = idx >> 5;
    const int ln = idx & 31;
    const int m  = ln & 15;
    const int kb = kk * 4 + ((ln >> 4) << 1);
    v2f a;
    if (m < 4) {
      const int k0 = m * 256 + kb;
      a.x = w2val(k0);
      a.y = w2val(k0 + 1);
    } else {
      a.x = 0.0f;
      a.y = 0.0f;
    }
    Afrag[idx] = a;
  }
  __syncthreads();   // only barrier; waves diverge freely afterwards

  const int Fw = wgi * FPG + wave * FPW;   // this wave's first frame/block
  if (Fw >= F_) return;                    // wave-uniform exit

  const float me = means[2 * b + 0];
  const float mt = means[2 * b + 1];

  float* eL = wbase + wave * WAVE_F;
  float* tL = eL + NBLK_W * ROWP;

  // ---- stage 4096 padded samples of e and t via async loads ----
  const long  s0 = (long)Fw * HOP_;
  const float* eg = est + (size_t)b * T_;
  const float* tg = tgt + (size_t)b * T_;
  for (int it = 0; it < 128; ++it) {
    const int o = it * 32 + lane;          // 0..4095 within span
    const int n = o >> 8;                  // block row
    const int r = o & 255;                 // within block
    const int j = reflect_idx(s0 + o);
    const uint32_t la_e = (uint32_t)(uintptr_t)&eL[n * ROWP + r];
    const uint32_t la_t = (uint32_t)(uintptr_t)&tL[n * ROWP + r];
    async_copy_b32(la_e, (unsigned long long)(uintptr_t)(eg + j));
    async_copy_b32(la_t, (unsigned long long)(uintptr_t)(tg + j));
  }
  wait_async0();

  // ---- taps-GEMM: 64 x (3 WMMA f32 16x16x4) accumulations ----
  // B 4x16 f32 layout assumed mirroring A: v0 = row K=kh, v1 = row K=kh+1,
  // N = lane&15 (per-lane column = block).
  v8f pd = {}, pt = {}, pe = {};
  const int m  = lane & 15;                       // N col (B) for this lane
  const int kh = (lane >> 4) << 1;                // 0 or 2: K offset for this lane half
  const v2f* Arow = Afrag + lane;                 // Afrag[kk*32 + lane]
  for (int kk = 0; kk < 64; ++kk) {
    const int off = m * ROWP + kk * 4 + kh;       // even -> 8B aligned
    const v2f A  = Arow[kk * 32];
    const v2f ev = *(const v2f*)(eL + off);
    const v2f tv = *(const v2f*)(tL + off);
    const float e0 = ev.x - me, e1 = ev.y - me;
    const float t0 = tv.x - mt, t1 = tv.y - mt;
    v2f Bd; Bd.x = e0 * t0; Bd.y = e1 * t1;
    v2f Bt; Bt.x = t0 * t0; Bt.y = t1 * t1;
    v2f Be; Be.x = e0 * e0; Be.y = e1 * e1;
    pd = __builtin_amdgcn_wmma_f32_16x16x4_f32(false, A, false, Bd, (short)0, pd, false, false);
    pt = __builtin_amdgcn_wmma_f32_16x16x4_f32(false, A, false, Bt, (short)0, pt, false, false);
    pe = __builtin_amdgcn_wmma_f32_16x16x4_f32(false, A, false, Be, (short)0, pe, false, false);
  }

  // C/D layout: VGPR v, lanes 0-15 -> row M=v, col N=lane. Rows 0..3 = taps.
  // Frame f (lane) needs P[j, f + j]: gather across lanes with shfl.
  float dotv = 0.f, ttv = 0.f, eev = 0.f;
#pragma unroll
  for (int jj = 0; jj < 4; ++jj) {
    dotv += __shfl(pd[jj], lane + jj);
    ttv  += __shfl(pt[jj], lane + jj);
    eev  += __shfl(pe[jj], lane + jj);
  }

  const int gf = Fw + lane;
  if (lane < FPW && gf < F_) {
    const float s   = ttv + EPS_;
    const float d2  = dotv * dotv;
    const float num = d2 * ttv / (s * s);               // ||scaled_target||^2
    const float den = eev - 2.f * d2 / s + num + EPS_;  // ||e_noise||^2 + eps
    const float ratio = num / den;
    out[(size_t)b * F_ + gf] = -10.f * __log10f(ratio + EPS_);
  }
}

// ---------------- host entry ----------------
extern "C" void kernel_launch(void* const* d_in, const int* in_sizes, int n_in,
                              void* d_out, int out_size, void* d_ws, size_t ws_size,
                              hipStream_t stream) {
  const float* est = (const float*)d_in[0];
  const float* tgt = (const float*)d_in[1];
  float* means = (float*)d_ws;      // 64 floats of scratch
  float* out   = (float*)d_out;     // [32, 1876]

  segsisdr_means<<<B_, 256, 0, stream>>>(est, tgt, means);

  const size_t shmem = (size_t)(2 * AFRAG_N + WPW * WAVE_F) * sizeof(float); // ~280 KB
  segsisdr_main<<<B_ * WGPB, 256, shmem, stream>>>(est, tgt, means, out);
}